// GAT_63840393888128
// MI455X (gfx1250) — compile-verified
//
#include <hip/hip_runtime.h>
#include <cstdint>

typedef float v2f __attribute__((ext_vector_type(2)));
typedef float v8f __attribute__((ext_vector_type(8)));

#define HEADS 8
#define NEG_SLOPE 0.2f
#define SM_EPS 1e-16f
#define NT 4   // column tiles per wave (64 output columns / wave)

// ---------------------------------------------------------------------------
// Dense GEMM  C[M,Ncols] = A[M,K] @ B[K,Ncols]  (all fp32, row-major)
// One wave32 computes a 16x(16*NT) strip of C via V_WMMA_F32_16X16X4_F32.
// Single-stage software pipeline: fragments for k+4 are loaded into separate
// registers before the k-step WMMAs consume the current fragments, so the
// scheduler can overlap the NT WMMAs with the next step's loads instead of
// draining loadcnt to 0 before each WMMA.
// A layout (ISA 7.12.2, 32-bit A 16x4): lanes 0-15 hold K=0,1; lanes 16-31 K=2,3.
// B layout: N = lane%16; lanes 0-15 hold K=0,1 in v0,v1; lanes 16-31 hold K=2,3.
// C layout: VGPR r -> row = r + 8*(lane>=16), col = lane%16.
// ---------------------------------------------------------------------------
__global__ void wmma_gemm_f32(const float* __restrict__ A,
                              const float* __restrict__ B,
                              float* __restrict__ C,
                              int M, int K, int Ncols) {
  int wave = (int)((blockIdx.x * blockDim.x + threadIdx.x) >> 5);
  int lane = (int)(threadIdx.x & 31);
  int tilesM  = (M + 15) >> 4;
  int groupsN = Ncols >> 6;                      // Ncols / (16*NT)
  if (wave >= tilesM * groupsN) return;          // whole wave exits together
  int tm = wave / groupsN;
  int tg = wave - tm * groupsN;
  int half = lane >> 4;                          // 0: K=0,1   1: K=2,3
  int l = lane & 15;

  int rowA = tm * 16 + l;
  if (rowA >= M) rowA = M - 1;                   // clamp (stores are guarded)
  const float* aP = A + (size_t)rowA * K + 2 * half;
  const float* bP = B + (size_t)(2 * half) * Ncols + tg * (16 * NT) + l;

  v8f acc[NT];
#pragma unroll
  for (int t = 0; t < NT; ++t) acc[t] = (v8f){};

  // prologue: fragments for k = 0..3
  v2f aCur, bCur[NT];
  aCur.x = aP[0];
  aCur.y = aP[1];
#pragma unroll
  for (int t = 0; t < NT; ++t) {
    bCur[t].x = bP[t * 16];
    bCur[t].y = bP[t * 16 + Ncols];
  }

  for (int k0 = 0; k0 < K - 4; k0 += 4) {
    aP += 4;
    bP += (size_t)4 * Ncols;
    // issue next step's loads into fresh registers
    v2f aNxt, bNxt[NT];
    aNxt.x = aP[0];
    aNxt.y = aP[1];
#pragma unroll
    for (int t = 0; t < NT; ++t) {
      bNxt[t].x = bP[t * 16];
      bNxt[t].y = bP[t * 16 + Ncols];
    }
    // consume current fragments (independent of the loads above)
#pragma unroll
    for (int t = 0; t < NT; ++t) {
      acc[t] = __builtin_amdgcn_wmma_f32_16x16x4_f32(false, aCur, false, bCur[t],
                                                     (short)0, acc[t],
                                                     false, false);
    }
    aCur = aNxt;
#pragma unroll
    for (int t = 0; t < NT; ++t) bCur[t] = bNxt[t];
  }
  // epilogue: last k-step
#pragma unroll
  for (int t = 0; t < NT; ++t) {
    acc[t] = __builtin_amdgcn_wmma_f32_16x16x4_f32(false, aCur, false, bCur[t],
                                                   (short)0, acc[t],
                                                   false, false);
  }

#pragma unroll
  for (int t = 0; t < NT; ++t) {
    int col = tg * (16 * NT) + t * 16 + l;
#pragma unroll
    for (int r = 0; r < 8; ++r) {
      int row = tm * 16 + r + 8 * half;
      if (row < M) C[(size_t)row * Ncols + col] = acc[t][r];
    }
  }
}

// ---------------------------------------------------------------------------
// a_src[n,h] = sum_c h[n,h,c]*att_src[h,c] ; a_dst likewise. One thread per (n,h).
// ---------------------------------------------------------------------------
__global__ void attn_coeff_kernel(const float* __restrict__ h,
                                  const float* __restrict__ att_src,
                                  const float* __restrict__ att_dst,
                                  float* __restrict__ asrc,
                                  float* __restrict__ adst,
                                  int N, int C) {
  long i = blockIdx.x * (long)blockDim.x + threadIdx.x;   // n*HEADS + head
  if (i >= (long)N * HEADS) return;
  int head = (int)(i & (HEADS - 1));
  const float* hp = h + (size_t)i * C;
  const float* as = att_src + (size_t)head * C;
  const float* ad = att_dst + (size_t)head * C;
  float s = 0.f, d = 0.f;
  for (int c = 0; c < C; ++c) {
    float v = hp[c];
    s += v * as[c];
    d += v * ad[c];
  }
  asrc[i] = s;
  adst[i] = d;
}

__global__ void fill_f32(float* __restrict__ p, float v, long n) {
  long i = blockIdx.x * (long)blockDim.x + threadIdx.x;
  if (i < n) p[i] = v;
}

__device__ __forceinline__ float lrelu(float v) {
  return v > 0.f ? v : NEG_SLOPE * v;
}

// ---------------------------------------------------------------------------
// Pass A: segment max of e = leakyrelu(a_src[s]+a_dst[d]) over dst.
// One thread per (edge, head); edges [0,E) from arrays, [E,E+N) are self-loops.
// ---------------------------------------------------------------------------
__global__ void edge_max_kernel(const int* __restrict__ src,
                                const int* __restrict__ dst,
                                int E, int N,
                                const float* __restrict__ asrc,
                                const float* __restrict__ adst,
                                float* __restrict__ m) {
  long i = blockIdx.x * (long)blockDim.x + threadIdx.x;
  long tot = (long)(E + N) * HEADS;
  if (i >= tot) return;
  int e = (int)(i >> 3);
  int head = (int)(i & 7);
  int s = (e < E) ? src[e] : (e - E);
  int d = (e < E) ? dst[e] : (e - E);
  float v = lrelu(asrc[(size_t)s * HEADS + head] + adst[(size_t)d * HEADS + head]);
  atomicMax(&m[(size_t)d * HEADS + head], v);
}

// Pass B: denom[d,h] += exp(e - m[d,h])
__global__ void edge_expsum_kernel(const int* __restrict__ src,
                                   const int* __restrict__ dst,
                                   int E, int N,
                                   const float* __restrict__ asrc,
                                   const float* __restrict__ adst,
                                   const float* __restrict__ m,
                                   float* __restrict__ denom) {
  long i = blockIdx.x * (long)blockDim.x + threadIdx.x;
  long tot = (long)(E + N) * HEADS;
  if (i >= tot) return;
  int e = (int)(i >> 3);
  int head = (int)(i & 7);
  int s = (e < E) ? src[e] : (e - E);
  int d = (e < E) ? dst[e] : (e - E);
  size_t dh = (size_t)d * HEADS + head;
  float v = lrelu(asrc[(size_t)s * HEADS + head] + adst[dh]);
  atomicAdd(&denom[dh], __expf(v - m[dh]));
}

// Pass C: agg[d,h,:] += alpha * h[s,h,:]   (float4 gathers, scalar atomics)
__global__ void edge_agg_kernel(const int* __restrict__ src,
                                const int* __restrict__ dst,
                                int E, int N, int C,
                                const float* __restrict__ asrc,
                                const float* __restrict__ adst,
                                const float* __restrict__ m,
                                const float* __restrict__ denom,
                                const float* __restrict__ h,
                                float* __restrict__ agg) {
  long i = blockIdx.x * (long)blockDim.x + threadIdx.x;
  long tot = (long)(E + N) * HEADS;
  if (i >= tot) return;
  int e = (int)(i >> 3);
  int head = (int)(i & 7);
  int s = (e < E) ? src[e] : (e - E);
  int d = (e < E) ? dst[e] : (e - E);
  size_t dh = (size_t)d * HEADS + head;
  float v = lrelu(asrc[(size_t)s * HEADS + head] + adst[dh]);
  float alpha = __expf(v - m[dh]) / (denom[dh] + SM_EPS);
  const float4* hs4 = (const float4*)(h + ((size_t)s * HEADS + head) * C);
  float* od = agg + dh * C;
  int C4 = C >> 2;
  for (int c4 = 0; c4 < C4; ++c4) {
    float4 hv = hs4[c4];
    atomicAdd(&od[c4 * 4 + 0], alpha * hv.x);
    atomicAdd(&od[c4 * 4 + 1], alpha * hv.y);
    atomicAdd(&od[c4 * 4 + 2], alpha * hv.z);
    atomicAdd(&od[c4 * 4 + 3], alpha * hv.w);
  }
}

// out[n,c] = mean_h agg[n,h,c] + bias[c], optional relu. One thread per (n,c).
__global__ void finalize_kernel(const float* __restrict__ agg,
                                const float* __restrict__ bias,
                                float* __restrict__ out,
                                int N, int C, int do_relu) {
  long i = blockIdx.x * (long)blockDim.x + threadIdx.x;
  if (i >= (long)N * C) return;
  int n = (int)(i / C);
  int c = (int)(i - (long)n * C);
  float s = 0.f;
#pragma unroll
  for (int hh = 0; hh < HEADS; ++hh)
    s += agg[((size_t)n * HEADS + hh) * C + c];
  s = s * (1.0f / HEADS) + bias[c];
  if (do_relu) s = s > 0.f ? s : 0.f;
  out[i] = s;
}

static inline unsigned nblk(long n, int b) { return (unsigned)((n + b - 1) / b); }

extern "C" void kernel_launch(void* const* d_in, const int* in_sizes, int n_in,
                              void* d_out, int out_size, void* d_ws, size_t ws_size,
                              hipStream_t stream) {
  const float* x        = (const float*)d_in[0];
  const int*   ei       = (const int*)  d_in[1];
  const float* W1       = (const float*)d_in[2];
  const float* att_src1 = (const float*)d_in[3];
  const float* att_dst1 = (const float*)d_in[4];
  const float* b1       = (const float*)d_in[5];
  const float* W2       = (const float*)d_in[6];
  const float* att_src2 = (const float*)d_in[7];
  const float* att_dst2 = (const float*)d_in[8];
  const float* b2       = (const float*)d_in[9];

  const int F_h   = in_sizes[3] / HEADS;             // 32
  const int F_out = in_sizes[7] / HEADS;             // 16
  const int F_in  = in_sizes[2] / (HEADS * F_h);     // 128
  const int N     = in_sizes[0] / F_in;              // 50000
  const int E     = in_sizes[1] / 2;                 // 800000
  const int C1    = HEADS * F_h;                     // 256
  const int C2    = HEADS * F_out;                   // 128
  const int* src = ei;
  const int* dst = ei + E;

  // Workspace layout (floats). h2/agg2 alias h1/agg1 regions after layer 1.
  float* ws = (float*)d_ws;
  size_t o = 0;
  float* h1   = ws + o; o += (size_t)N * C1;     // layer1 features; later h2
  float* agg  = ws + o; o += (size_t)N * C1;     // agg1; later agg2
  float* asrc = ws + o; o += (size_t)N * HEADS;
  float* adst = ws + o; o += (size_t)N * HEADS;
  float* mbuf = ws + o; o += (size_t)N * HEADS;
  float* dbuf = ws + o; o += (size_t)N * HEADS;
  float* x2   = ws + o; o += (size_t)N * F_h;
  float* h2   = h1;                               // N*C2 <= N*C1
  float* outp = (float*)d_out;

  const int TB = 256;
  const long NH = (long)N * HEADS;
  const long ETOT = (long)(E + N) * HEADS;

  // ---------------- Layer 1 ----------------
  {
    long waves = (long)((N + 15) / 16) * (C1 / (16 * NT));
    wmma_gemm_f32<<<nblk(waves * 32, TB), TB, 0, stream>>>(x, W1, h1, N, F_in, C1);
  }
  attn_coeff_kernel<<<nblk(NH, TB), TB, 0, stream>>>(h1, att_src1, att_dst1,
                                                     asrc, adst, N, F_h);
  fill_f32<<<nblk(NH, TB), TB, 0, stream>>>(mbuf, -3.0e38f, NH);
  fill_f32<<<nblk(NH, TB), TB, 0, stream>>>(dbuf, 0.0f, NH);
  fill_f32<<<nblk((long)N * C1, TB), TB, 0, stream>>>(agg, 0.0f, (long)N * C1);
  edge_max_kernel<<<nblk(ETOT, TB), TB, 0, stream>>>(src, dst, E, N, asrc, adst, mbuf);
  edge_expsum_kernel<<<nblk(ETOT, TB), TB, 0, stream>>>(src, dst, E, N, asrc, adst,
                                                        mbuf, dbuf);
  edge_agg_kernel<<<nblk(ETOT, TB), TB, 0, stream>>>(src, dst, E, N, F_h, asrc, adst,
                                                     mbuf, dbuf, h1, agg);
  finalize_kernel<<<nblk((long)N * F_h, TB), TB, 0, stream>>>(agg, b1, x2, N, F_h, 1);

  // ---------------- Layer 2 ----------------
  {
    long waves = (long)((N + 15) / 16) * (C2 / (16 * NT));
    wmma_gemm_f32<<<nblk(waves * 32, TB), TB, 0, stream>>>(x2, W2, h2, N, F_h, C2);
  }
  attn_coeff_kernel<<<nblk(NH, TB), TB, 0, stream>>>(h2, att_src2, att_dst2,
                                                     asrc, adst, N, F_out);
  fill_f32<<<nblk(NH, TB), TB, 0, stream>>>(mbuf, -3.0e38f, NH);
  fill_f32<<<nblk(NH, TB), TB, 0, stream>>>(dbuf, 0.0f, NH);
  fill_f32<<<nblk((long)N * C2, TB), TB, 0, stream>>>(agg, 0.0f, (long)N * C2);
  edge_max_kernel<<<nblk(ETOT, TB), TB, 0, stream>>>(src, dst, E, N, asrc, adst, mbuf);
  edge_expsum_kernel<<<nblk(ETOT, TB), TB, 0, stream>>>(src, dst, E, N, asrc, adst,
                                                        mbuf, dbuf);
  edge_agg_kernel<<<nblk(ETOT, TB), TB, 0, stream>>>(src, dst, E, N, F_out, asrc, adst,
                                                     mbuf, dbuf, h2, agg);
  finalize_kernel<<<nblk((long)N * F_out, TB), TB, 0, stream>>>(agg, b2, outp,
                                                                N, F_out, 0);
}